// ScaledDotProductAttentionWithGoc_71433896067278
// MI455X (gfx1250) — compile-verified
//
#include <hip/hip_runtime.h>
#include <cstdint>
#include <cstddef>

// ---------------- problem constants (from reference) ----------------
constexpr int CB   = 256;   // batch
constexpr int CNQ  = 256;   // queries per batch
constexpr int CD   = 512;   // model dim
constexpr int CH   = 8;     // heads
constexpr int CNK  = 99;    // enc keys
constexpr int CNEW = 50;    // new region
constexpr int CGO  = 49;    // NK - NEW
constexpr float CSCALE = 0.125f;  // 1/sqrt(64)

// ---------------- WMMA helpers (CDNA5, wave32) ----------------
typedef float v2f __attribute__((ext_vector_type(2)));
typedef float v8f __attribute__((ext_vector_type(8)));
typedef unsigned int v4u __attribute__((ext_vector_type(4)));
typedef int v4i __attribute__((ext_vector_type(4)));
typedef int v8i __attribute__((ext_vector_type(8)));

__device__ __forceinline__ v8f wmma4(v2f a, v2f b, v8f c) {
  // D = A(16x4,f32) * B(4x16,f32) + C(16x16,f32)
  return __builtin_amdgcn_wmma_f32_16x16x4_f32(false, a, false, b, (short)0, c,
                                               false, false);
}
__device__ __forceinline__ v8f vzero8() { v8f z = {}; return z; }

// ---------------- Tensor Data Mover: 2D tile -> padded LDS ----------------
// D# packing per CDNA5 ISA.
// tile_w/tile_h: tile extent in 4B elements/rows. tdim0/tdim1: valid extent
// from the tile origin (rows/cols beyond are hardware zero-filled).
// row_stride: elements between rows. padi/pada: LDS pad codes
// (interval = 2^(padi+1) DWORDs, amount = pada+1 DWORDs).
__device__ __forceinline__ void tdm_load_2d(const void* gaddr,
                                            unsigned lds_off_bytes,
                                            unsigned tile_w, unsigned tile_h,
                                            unsigned tdim0, unsigned tdim1,
                                            unsigned row_stride,
                                            unsigned padi, unsigned pada) {
  const unsigned long long ga = (unsigned long long)gaddr;
  v4u g0;
  g0.x = 1u;                                      // count=1, user descriptor
  g0.y = lds_off_bytes;                           // lds_addr
  g0.z = (unsigned)ga;                            // global_addr[31:0]
  g0.w = ((unsigned)(ga >> 32) & 0x01FFFFFFu)     // global_addr[56:32]
         | 0x80000000u;                           // type=2 ("image")
  v8i g1;
  g1[0] = (int)((2u << 16)                        // data_size = 4B
                | (1u << 20)                      // pad_enable
                | (padi << 22) | (pada << 25));
  g1[1] = (int)((tdim0 & 0xFFFFu) << 16);         // tensor_dim0[15:0]
  g1[2] = (int)(((tdim0 >> 16) & 0xFFFFu) | ((tdim1 & 0xFFFFu) << 16));
  g1[3] = (int)(((tdim1 >> 16) & 0xFFFFu) | (tile_w << 16));  // tile_dim0
  g1[4] = (int)(tile_h & 0xFFFFu);                // tile_dim1 (tile_dim2=0)
  g1[5] = (int)row_stride;                        // tensor_dim0_stride[31:0]
  g1[6] = 0;                                      // stride hi / dim1_stride lo
  g1[7] = 0;
  const v4i z4 = {0, 0, 0, 0};                    // groups 2/3: 2D tensor
  const v8i z8 = {0, 0, 0, 0, 0, 0, 0, 0};        // trailing group (clang-23 ABI)
  __builtin_amdgcn_tensor_load_to_lds(g0, g1, z4, z4, z8, 0);
}
__device__ __forceinline__ unsigned lds_off(const void* p) {
  return (unsigned)(size_t)p;  // low 32 bits of shared aperture = LDS offset
}

__device__ __forceinline__ void atomic_min_f32(float* addr, float val) {
  unsigned int* ua = reinterpret_cast<unsigned int*>(addr);
  unsigned int old = __float_as_uint(*addr);
  for (;;) {
    float cur = __uint_as_float(old);
    if (!(val < cur)) break;
    unsigned int assumed = old;
    old = atomicCAS(ua, assumed, __float_as_uint(val));
    if (old == assumed) break;
  }
}

__global__ void init_mins_kernel(float* mins) {
  if (threadIdx.x < 2) mins[threadIdx.x] = __builtin_inff();
}

// ---------------- generic projection GEMM: Y = X @ W^T + b ----------------
// X rows are mapped (batch,seq) -> X + (batch*batch_stride + seq)*512.
// Tile: 128(M) x 64(N), K-step 16, 8 waves, each wave 32x32 = 2x2 WMMA tiles.
// W tile is staged by the Tensor Data Mover (dense 64x16, stride 512,
// hardware-padded to [64][20]); A tile uses vector loads (rows cross batch
// boundaries and need the M guard).
__global__ __launch_bounds__(256) void proj_gemm_kernel(
    const float* __restrict__ X, const float* __restrict__ W,
    const float* __restrict__ bias, float* __restrict__ Y,
    int seq_len, int batch_stride, int M_total) {
  __shared__ float As[128 * 20];  // [m][k], row pad 20 (80B, 16B-aligned rows)
  __shared__ float Bt[64 * 20];   // [n][k], filled by TDM (16 + 4 DWORD pad)
  const int t = threadIdx.x;
  const int lane = t & 31;
  const int wave = t >> 5;
  const int wm = wave & 3;   // 4 row-blocks of 32
  const int wn = wave >> 2;  // 2 col-blocks of 32
  const int m0 = blockIdx.x * 128;
  const int n0 = blockIdx.y * 64;

  v8f acc[2][2];
#pragma unroll
  for (int i = 0; i < 2; ++i)
#pragma unroll
    for (int j = 0; j < 2; ++j) acc[i][j] = vzero8();

  for (int kt = 0; kt < 512; kt += 16) {
    // W tile via TDM: issued once per workgroup (wave 0), overlapped with the
    // A-tile vector loads below; completion gated by s_wait_tensorcnt.
    if (wave == 0) {
      tdm_load_2d(W + (size_t)n0 * 512 + kt, lds_off(Bt),
                  /*tile_w=*/16, /*tile_h=*/64, /*tdim0=*/16, /*tdim1=*/64,
                  /*stride=*/512, /*padi=*/3 /*16 DW*/, /*pada=*/3 /*4 DW*/);
    }
    // A tile: 128 rows x 16 cols = 512 float4, 2 per thread
#pragma unroll
    for (int q = 0; q < 2; ++q) {
      const int f = t + q * 256;
      const int row = f >> 2, c4 = f & 3;
      const int mg = m0 + row;
      float4 val = make_float4(0.f, 0.f, 0.f, 0.f);
      if (mg < M_total) {
        const int bb = mg / seq_len;
        const int ss = mg - bb * seq_len;
        const float* src = X + ((size_t)bb * batch_stride + ss) * 512 + kt + (c4 << 2);
        val = *(const float4*)src;
        if (kt + 16 < 512) __builtin_prefetch(src + 16, 0, 1);
      }
      *(float4*)&As[row * 20 + (c4 << 2)] = val;
    }
    __builtin_amdgcn_s_wait_tensorcnt((short)0);  // no-op for waves 1..7
    __syncthreads();
#pragma unroll
    for (int ks = 0; ks < 4; ++ks) {
      const int kb = (ks << 2) + ((lane >> 4) << 1);
      const int cb = lane & 15;
      v2f a0 = *(const v2f*)&As[(wm * 32 + cb) * 20 + kb];
      v2f a1 = *(const v2f*)&As[(wm * 32 + 16 + cb) * 20 + kb];
      v2f b0 = *(const v2f*)&Bt[(wn * 32 + cb) * 20 + kb];
      v2f b1 = *(const v2f*)&Bt[(wn * 32 + 16 + cb) * 20 + kb];
      acc[0][0] = wmma4(a0, b0, acc[0][0]);
      acc[0][1] = wmma4(a0, b1, acc[0][1]);
      acc[1][0] = wmma4(a1, b0, acc[1][0]);
      acc[1][1] = wmma4(a1, b1, acc[1][1]);
    }
    __syncthreads();
  }
#pragma unroll
  for (int i = 0; i < 2; ++i)
#pragma unroll
    for (int j = 0; j < 2; ++j) {
      const int col = n0 + wn * 32 + j * 16 + (lane & 15);
      const float bv = bias[col];
#pragma unroll
      for (int r = 0; r < 8; ++r) {
        const int m = m0 + wm * 32 + i * 16 + r + ((lane >> 4) << 3);
        if (m < M_total) Y[(size_t)m * 512 + col] = acc[i][j][r] + bv;
      }
    }
}

// ---------------- phase 1: global mins of masked att / new_att ----------------
__global__ __launch_bounds__(256) void attn_min_kernel(
    const float* __restrict__ qp, const float* __restrict__ kp,
    const float* __restrict__ nkp, const unsigned char* __restrict__ amask,
    float* __restrict__ mins) {
  extern __shared__ float sm[];
  float* qs  = sm;               // [256][68]
  float* ksm = qs + 256 * 68;    // [112][68]  rows >= 99 HW zero-filled
  float* nsm = ksm + 112 * 68;   // [64][68]   rows >= 50 HW zero-filled
  float* mk  = nsm + 64 * 68;    // [112]      1.0 = masked
  __shared__ float red0[256];
  __shared__ float red1[256];
  const int t = threadIdx.x, lane = t & 31, wave = t >> 5;
  const int bh = blockIdx.x, b = bh >> 3, h = bh & 7;

  // ---- TDM staging: every wave DMAs its own 32-row q chunk; waves 0/1 also
  // bring K and NK tiles with hardware OOB zero-fill past the valid rows.
  tdm_load_2d(qp + ((size_t)(b * 256 + wave * 32)) * 512 + h * 64,
              lds_off(&qs[wave * 32 * 68]),
              64, 32, 64, 32, 512, /*padi=*/5 /*64 DW*/, /*pada=*/3 /*4 DW*/);
  if (wave == 0)
    tdm_load_2d(kp + ((size_t)b * CNK) * 512 + h * 64, lds_off(ksm),
                64, 112, 64, CNK, 512, 5, 3);
  if (wave == 1)
    tdm_load_2d(nkp + ((size_t)b * CNEW) * 512 + h * 64, lds_off(nsm),
                64, 64, 64, CNEW, 512, 5, 3);
  if (t < 112) mk[t] = (t < CNK) ? (amask[b * CNK + t] ? 1.f : 0.f) : 1.f;
  __builtin_amdgcn_s_wait_tensorcnt((short)0);
  __syncthreads();

  const int row0 = wave * 32;
  const int cb = lane & 15;

  float lmin = __builtin_inff();
  {
    v8f acc[2][7];
#pragma unroll
    for (int i = 0; i < 2; ++i)
#pragma unroll
      for (int j = 0; j < 7; ++j) acc[i][j] = vzero8();
#pragma unroll 4
    for (int ki = 0; ki < 16; ++ki) {
      const int kb = (ki << 2) + ((lane >> 4) << 1);
      v2f a0 = *(const v2f*)&qs[(row0 + cb) * 68 + kb];
      v2f a1 = *(const v2f*)&qs[(row0 + 16 + cb) * 68 + kb];
#pragma unroll
      for (int j = 0; j < 7; ++j) {
        v2f bf = *(const v2f*)&ksm[(j * 16 + cb) * 68 + kb];
        acc[0][j] = wmma4(a0, bf, acc[0][j]);
        acc[1][j] = wmma4(a1, bf, acc[1][j]);
      }
    }
#pragma unroll
    for (int j = 0; j < 7; ++j) {
      const int col = j * 16 + cb;
      if (col < CNK) {
        const float mv = mk[col];
#pragma unroll
        for (int i = 0; i < 2; ++i)
#pragma unroll
          for (int r = 0; r < 8; ++r) {
            const float v = (mv != 0.f) ? 0.f : acc[i][j][r] * CSCALE;
            lmin = fminf(lmin, v);
          }
      }
    }
  }
  float lmin2 = __builtin_inff();
  {
    v8f acn[2][4];
#pragma unroll
    for (int i = 0; i < 2; ++i)
#pragma unroll
      for (int j = 0; j < 4; ++j) acn[i][j] = vzero8();
#pragma unroll 4
    for (int ki = 0; ki < 16; ++ki) {
      const int kb = (ki << 2) + ((lane >> 4) << 1);
      v2f a0 = *(const v2f*)&qs[(row0 + cb) * 68 + kb];
      v2f a1 = *(const v2f*)&qs[(row0 + 16 + cb) * 68 + kb];
#pragma unroll
      for (int j = 0; j < 4; ++j) {
        v2f bf = *(const v2f*)&nsm[(j * 16 + cb) * 68 + kb];
        acn[0][j] = wmma4(a0, bf, acn[0][j]);
        acn[1][j] = wmma4(a1, bf, acn[1][j]);
      }
    }
#pragma unroll
    for (int j = 0; j < 4; ++j) {
      const int col = j * 16 + cb;
      if (col < CNEW) {
        const float mv = mk[col];
#pragma unroll
        for (int i = 0; i < 2; ++i)
#pragma unroll
          for (int r = 0; r < 8; ++r) {
            const float v = (mv != 0.f) ? 0.f : acn[i][j][r] * CSCALE;
            lmin2 = fminf(lmin2, v);
          }
      }
    }
  }
  red0[t] = lmin;
  red1[t] = lmin2;
  __syncthreads();
  for (int s = 128; s > 0; s >>= 1) {
    if (t < s) {
      red0[t] = fminf(red0[t], red0[t + s]);
      red1[t] = fminf(red1[t], red1[t + s]);
    }
    __syncthreads();
  }
  if (t == 0) {
    atomic_min_f32(&mins[0], red0[0]);
    atomic_min_f32(&mins[1], red1[0]);
  }
}

// ---------------- phase 2: full attention per (b,h), LDS-resident ----------------
__global__ __launch_bounds__(256) void attn_ctx_kernel(
    const float* __restrict__ qp, const float* __restrict__ kp,
    const float* __restrict__ nkp, const float* __restrict__ vp,
    const float* __restrict__ aligns, const unsigned char* __restrict__ amask,
    const float* __restrict__ mins, float* __restrict__ ctx) {
  extern __shared__ float sm[];
  float* qs   = sm;                  // [256][68]  (aliased by snew after use)
  float* snew = sm;                  // [256][54]
  float* ksm  = sm + 256 * 68;       // [112][68]  rows >= 99 HW zero-filled
  float* nsm  = ksm + 112 * 68;      // [64][68]   rows >= 50 HW zero-filled
  float* vsm  = nsm + 64 * 68;       // [100][68]  row 99 HW zero-filled
  float* alm  = vsm + 100 * 68;      // [52][67]   zero-padded aligns
  float* S    = alm + 52 * 67;       // [256][114] scores -> probabilities
  float* mk   = S + 256 * 114;       // [112]

  const int t = threadIdx.x, lane = t & 31, wave = t >> 5;
  const int bh = blockIdx.x, b = bh >> 3, h = bh & 7;

  // ---- TDM staging (pad 64+4 DWORDs per row -> [*][68] layout) ----
  tdm_load_2d(qp + ((size_t)(b * 256 + wave * 32)) * 512 + h * 64,
              lds_off(&qs[wave * 32 * 68]),
              64, 32, 64, 32, 512, 5, 3);
  if (wave == 0)
    tdm_load_2d(kp + ((size_t)b * CNK) * 512 + h * 64, lds_off(ksm),
                64, 112, 64, CNK, 512, 5, 3);
  if (wave == 1)
    tdm_load_2d(nkp + ((size_t)b * CNEW) * 512 + h * 64, lds_off(nsm),
                64, 64, 64, CNEW, 512, 5, 3);
  if (wave == 2)
    tdm_load_2d(vp + ((size_t)b * CNK) * 512 + h * 64, lds_off(vsm),
                64, 100, 64, CNK, 512, 5, 3);
  // aligns (49-float rows: not expressible as power-of-2 pad) + mask: vector
  for (int f = t; f < 52 * 67; f += 256) {
    const int r = f / 67, c = f - r * 67;
    alm[f] = (r < CNEW && c < CGO) ? aligns[(size_t)b * (CNEW * CGO) + r * CGO + c] : 0.f;
  }
  if (t < 112) mk[t] = (t < CNK) ? (amask[b * CNK + t] ? 1.f : 0.f) : 1.f;
  __builtin_amdgcn_s_wait_tensorcnt((short)0);
  __syncthreads();

  const int row0 = wave * 32;
  const int cb = lane & 15;
  const int rofs = (lane >> 4) << 3;
  const float sh_att = fminf(mins[0], 0.f);
  const float sh_new = fminf(mins[1], 0.f);

  // ---- att = q@k^T and new_att = q@nk^T kept in registers ----
  v8f acc[2][7];
  v8f acn[2][4];
#pragma unroll
  for (int i = 0; i < 2; ++i) {
#pragma unroll
    for (int j = 0; j < 7; ++j) acc[i][j] = vzero8();
#pragma unroll
    for (int j = 0; j < 4; ++j) acn[i][j] = vzero8();
  }
#pragma unroll 2
  for (int ki = 0; ki < 16; ++ki) {
    const int kb = (ki << 2) + ((lane >> 4) << 1);
    v2f a0 = *(const v2f*)&qs[(row0 + cb) * 68 + kb];
    v2f a1 = *(const v2f*)&qs[(row0 + 16 + cb) * 68 + kb];
#pragma unroll
    for (int j = 0; j < 7; ++j) {
      v2f bf = *(const v2f*)&ksm[(j * 16 + cb) * 68 + kb];
      acc[0][j] = wmma4(a0, bf, acc[0][j]);
      acc[1][j] = wmma4(a1, bf, acc[1][j]);
    }
#pragma unroll
    for (int j = 0; j < 4; ++j) {
      v2f bf = *(const v2f*)&nsm[(j * 16 + cb) * 68 + kb];
      acn[0][j] = wmma4(a0, bf, acn[0][j]);
      acn[1][j] = wmma4(a1, bf, acn[1][j]);
    }
  }
  __syncthreads();  // all q reads done -> qs region reusable as snew

  // ---- write shifted+masked scores ----
#pragma unroll
  for (int j = 0; j < 7; ++j) {
    const int col = j * 16 + cb;
    const float mv = mk[col];
#pragma unroll
    for (int i = 0; i < 2; ++i)
#pragma unroll
      for (int r = 0; r < 8; ++r) {
        const int row = row0 + i * 16 + r + rofs;
        S[row * 114 + col] = (mv != 0.f) ? 0.f : (acc[i][j][r] * CSCALE - sh_att);
      }
  }
#pragma unroll
  for (int j = 0; j < 4; ++j) {
    const int col = j * 16 + cb;
#pragma unroll
    for (int i = 0; i < 2; ++i)
#pragma unroll
      for (int r = 0; r < 8; ++r) {
        const int row = row0 + i * 16 + r + rofs;
        float v = 0.f;
        if (col < CNEW && mk[col] == 0.f) v = acn[i][j][r] * CSCALE - sh_new;
        if (col < 54) snew[row * 54 + col] = v;
      }
  }
  __syncthreads();

  // ---- goatt = (new_att @ aligns)/25 added into S[:, 50:99] ----
  v8f gac[2][4];
#pragma unroll
  for (int i = 0; i < 2; ++i)
#pragma unroll
    for (int j = 0; j < 4; ++j) gac[i][j] = vzero8();
#pragma unroll 1
  for (int ki = 0; ki < 13; ++ki) {  // K = 52 (zero-padded past 50)
    const int kb = (ki << 2) + ((lane >> 4) << 1);
    v2f a0 = *(const v2f*)&snew[(row0 + cb) * 54 + kb];
    v2f a1 = *(const v2f*)&snew[(row0 + 16 + cb) * 54 + kb];
#pragma unroll
    for (int j = 0; j < 4; ++j) {
      const int g = j * 16 + cb;
      v2f bf;
      bf.x = alm[kb * 67 + g];
      bf.y = alm[(kb + 1) * 67 + g];
      gac[0][j] = wmma4(a0, bf, gac[0][j]);
      gac[1][j] = wmma4(a1, bf, gac[1][j]);
    }
  }
#pragma unroll
  for (int j = 0; j < 4; ++j) {
    const int g = j * 16 + cb;
    if (g < CGO) {
#pragma unroll
      for (int i = 0; i < 2; ++i)
#pragma unroll
        for (int r = 0; r < 8; ++r) {
          const int row = row0 + i * 16 + r + rofs;
          S[row * 114 + (CNEW + g)] += gac[i][j][r] * 0.04f;  // /25
        }
    }
  }
  __syncthreads();

  // ---- masked softmax, one row per thread ----
  {
    const int row = t;
    float mx = -__builtin_inff();
    for (int c = 0; c < CNK; ++c)
      if (mk[c] == 0.f) mx = fmaxf(mx, S[row * 114 + c]);
    float sum = 0.f;
    for (int c = 0; c < CNK; ++c) {
      const float p = (mk[c] != 0.f) ? 0.f : __expf(S[row * 114 + c] - mx);
      S[row * 114 + c] = p;
      sum += p;
    }
    const float inv = (sum > 0.f) ? (1.f / sum) : 0.f;
    for (int c = 0; c < CNK; ++c) S[row * 114 + c] *= inv;
    S[row * 114 + 99] = 0.f;  // zero the K-pad column read by the last k-step
  }
  __syncthreads();

  // ---- out = P @ V ----
  v8f oac[2][4];
#pragma unroll
  for (int i = 0; i < 2; ++i)
#pragma unroll
    for (int j = 0; j < 4; ++j) oac[i][j] = vzero8();
#pragma unroll 1
  for (int ki = 0; ki < 25; ++ki) {  // K = 100 (row 99 of V and col 99 of P are 0)
    const int kb = (ki << 2) + ((lane >> 4) << 1);
    v2f a0 = *(const v2f*)&S[(row0 + cb) * 114 + kb];
    v2f a1 = *(const v2f*)&S[(row0 + 16 + cb) * 114 + kb];
#pragma unroll
    for (int j = 0; j < 4; ++j) {
      const int n = j * 16 + cb;
      v2f bf;
      bf.x = vsm[kb * 68 + n];
      bf.y = vsm[(kb + 1) * 68 + n];
      oac[0][j] = wmma4(a0, bf, oac[0][j]);
      oac[1][j] = wmma4(a1, bf, oac[1][j]);
    }
  }
#pragma unroll
  for (int j = 0; j < 4; ++j) {
    const int col = j * 16 + cb;
#pragma unroll
    for (int i = 0; i < 2; ++i)
#pragma unroll
      for (int r = 0; r < 8; ++r) {
        const int row = row0 + i * 16 + r + rofs;
        ctx[((size_t)(b * 256 + row)) * 512 + h * 64 + col] = oac[i][j][r];
      }
  }
}

// ---------------- host-side orchestration ----------------
extern "C" void kernel_launch(void* const* d_in, const int* in_sizes, int n_in,
                              void* d_out, int out_size, void* d_ws, size_t ws_size,
                              hipStream_t stream) {
  (void)in_sizes; (void)n_in; (void)out_size; (void)ws_size;
  const float* queries = (const float*)d_in[0];
  const float* keys    = (const float*)d_in[1];
  const float* values  = (const float*)d_in[2];
  const unsigned char* amask = (const unsigned char*)d_in[3];  // bool, 1B/elem
  const float* aligns  = (const float*)d_in[4];
  const float* Wq  = (const float*)d_in[5];
  const float* bq  = (const float*)d_in[6];
  const float* Wk  = (const float*)d_in[7];
  const float* bk  = (const float*)d_in[8];
  const float* Wv  = (const float*)d_in[9];
  const float* bv  = (const float*)d_in[10];
  const float* Wnk = (const float*)d_in[11];
  const float* bnk = (const float*)d_in[12];
  const float* Wo  = (const float*)d_in[13];
  const float* bo  = (const float*)d_in[14];
  float* out = (float*)d_out;
  float* ws  = (float*)d_ws;

  // workspace layout (floats): q, k, v, nk projections, context, 2 mins (~399 MB)
  float* qp   = ws;
  float* kpp  = qp + (size_t)CB * CNQ * CD;
  float* vpp  = kpp + (size_t)CB * CNK * CD;
  float* nkp  = vpp + (size_t)CB * CNK * CD;
  float* ctx  = nkp + (size_t)CB * CNEW * CD;
  float* mins = ctx + (size_t)CB * CNQ * CD;

  const size_t smemB = (size_t)(256 * 68 + 112 * 68 + 64 * 68 + 112) * sizeof(float);
  const size_t smemC = (size_t)(256 * 68 + 112 * 68 + 64 * 68 + 100 * 68 +
                                52 * 67 + 256 * 114 + 112) * sizeof(float);
  // allow >64KB dynamic LDS (CDNA5: up to 320KB/WG); idempotent, capture-safe
  (void)hipFuncSetAttribute((const void*)attn_min_kernel,
                            hipFuncAttributeMaxDynamicSharedMemorySize, (int)smemB);
  (void)hipFuncSetAttribute((const void*)attn_ctx_kernel,
                            hipFuncAttributeMaxDynamicSharedMemorySize, (int)smemC);

  const dim3 blk(256, 1, 1);
  hipLaunchKernelGGL(init_mins_kernel, dim3(1), dim3(32), 0, stream, mins);

  // projections: Y = X @ W^T + b
  hipLaunchKernelGGL(proj_gemm_kernel, dim3(512, 8), blk, 0, stream,
                     queries, Wq, bq, qp, CNQ, CNQ, CB * CNQ);
  hipLaunchKernelGGL(proj_gemm_kernel, dim3(198, 8), blk, 0, stream,
                     keys, Wk, bk, kpp, CNK, CNK + CNEW, CB * CNK);
  hipLaunchKernelGGL(proj_gemm_kernel, dim3(198, 8), blk, 0, stream,
                     values, Wv, bv, vpp, CNK, CNK, CB * CNK);
  hipLaunchKernelGGL(proj_gemm_kernel, dim3(100, 8), blk, 0, stream,
                     keys + (size_t)CNK * CD, Wnk, bnk, nkp, CNEW, CNK + CNEW, CB * CNEW);

  // phase 1: global minima of masked score tensors
  hipLaunchKernelGGL(attn_min_kernel, dim3(CB * CH), blk, smemB, stream,
                     qp, kpp, nkp, amask, mins);
  // phase 2: shift, goatt, softmax, P@V -> context
  hipLaunchKernelGGL(attn_ctx_kernel, dim3(CB * CH), blk, smemC, stream,
                     qp, kpp, nkp, vpp, aligns, amask, mins, ctx);

  // output projection
  hipLaunchKernelGGL(proj_gemm_kernel, dim3(512, 8), blk, 0, stream,
                     ctx, Wo, bo, out, CNQ, CNQ, CB * CNQ);
}